// SingleScaleFixedSizeDeformAttnOnnx_59983513256472
// MI455X (gfx1250) — compile-verified
//
#include <hip/hip_runtime.h>

// Problem constants (fixed by the reference).
#define BS   4
#define NQ   10000
#define NH   8
#define HD   32
#define NP   4
#define H_   100
#define W_   100
#define HW_  (H_ * W_)
#define QT   16                 // queries per wave tile (WMMA M dim)
#define NTQ  (NQ / QT)          // 625 query tiles
#define PIXSTRIDE (NH * HD)     // 256 floats between pixels in `value`

typedef float v2f __attribute__((ext_vector_type(2)));
typedef float v8f __attribute__((ext_vector_type(8)));

// One wave (32 lanes) per (b, h, 16-query tile).
// Lane roles: n = lane&15 -> channel column (and WMMA row selector for A),
//             hi = lane>>4 -> which half of the K=4 point set this lane feeds.
// Per iteration t (query q0+t):
//   B (4x16 f32):  vgpr0 = sampled[point pa][n], vgpr1 = sampled[point pa+1][n]
//                  (pa = 0 for lanes 0-15 -> K rows 0/1; pa = 2 for lanes 16-31 -> K rows 2/3)
//   A (16x4 f32):  row t = attention weights of query t, other rows 0.
//   D (16x16 f32): accumulates; D[m, n] = output of query q0+m, channel n (+16 for hi tile).
__global__ __launch_bounds__(256)
void deform_attn_wmma_kernel(const float* __restrict__ value,
                             const float* __restrict__ loc,
                             const float* __restrict__ aw,
                             float* __restrict__ out) {
  const int lane = threadIdx.x & 31;
  const int wave = threadIdx.x >> 5;
  const int tile = blockIdx.x * 8 + wave;        // 20000 tiles, 8 waves/block

  const int b   = tile / (NH * NTQ);
  const int rem = tile - b * (NH * NTQ);
  const int h   = rem / NTQ;
  const int qt  = rem - h * NTQ;
  const int q0  = qt * QT;

  const int n  = lane & 15;                      // channel column / A-row selector
  const int hi = lane >> 4;                      // 0: points 0,1   1: points 2,3
  const int pa = hi << 1;

  // value[((b*HW + pix)*NH + h)*HD + d]  ->  vb + pix*PIXSTRIDE + d
  const float* vb   = value + (size_t)b * HW_ * PIXSTRIDE + h * HD;
  // loc[(((b*NQ+q)*NH + h)*NP + p)*2]
  const float* locb = loc + ((size_t)b * NQ * NH + h) * (NP * 2);
  // aw[((b*NQ+q)*NH + h)*NP + p]
  const float* awb  = aw + ((size_t)b * NQ * NH + h) * NP;

  v8f dlo = {};   // channels 0..15
  v8f dhi = {};   // channels 16..31

  for (int t = 0; t < QT; ++t) {
    const int q = q0 + t;

    // Wave-uniform per-point metadata (broadcast loads): this lane's two points.
    const float4 l4 = *(const float4*)(locb + (size_t)q * (NH * NP * 2) + pa * 2);
    const float2 w2 = *(const float2*)(awb  + (size_t)q * (NH * NP)     + pa);

    if (t + 1 < QT) {  // gfx1250 global_prefetch_b8 on next query's metadata
      __builtin_prefetch(locb + (size_t)(q + 1) * (NH * NP * 2), 0, 0);
      __builtin_prefetch(awb  + (size_t)(q + 1) * (NH * NP), 0, 0);
    }

    float slo[2], shi[2];
    #pragma unroll
    for (int j = 0; j < 2; ++j) {
      const float lx = (j == 0) ? l4.x : l4.z;
      const float ly = (j == 0) ? l4.y : l4.w;
      // grid = 2*loc-1; unnormalize (align_corners=False): x = loc*W - 0.5
      const float x = lx * (float)W_ - 0.5f;
      const float y = ly * (float)H_ - 0.5f;
      const float xf = floorf(x), yf = floorf(y);
      const float wx1 = x - xf, wy1 = y - yf;
      const float wx0 = 1.0f - wx1, wy0 = 1.0f - wy1;
      const int x0 = (int)xf, y0 = (int)yf;
      const int x1 = x0 + 1,  y1 = y0 + 1;
      const bool vx0 = (unsigned)x0 < (unsigned)W_;
      const bool vx1 = (unsigned)x1 < (unsigned)W_;
      const bool vy0 = (unsigned)y0 < (unsigned)H_;
      const bool vy1 = (unsigned)y1 < (unsigned)H_;
      const float w00 = (vx0 && vy0) ? wx0 * wy0 : 0.0f;
      const float w10 = (vx1 && vy0) ? wx1 * wy0 : 0.0f;
      const float w01 = (vx0 && vy1) ? wx0 * wy1 : 0.0f;
      const float w11 = (vx1 && vy1) ? wx1 * wy1 : 0.0f;
      const int xc0 = x0 < 0 ? 0 : (x0 > W_ - 1 ? W_ - 1 : x0);
      const int xc1 = x1 < 0 ? 0 : (x1 > W_ - 1 ? W_ - 1 : x1);
      const int yc0 = y0 < 0 ? 0 : (y0 > H_ - 1 ? H_ - 1 : y0);
      const int yc1 = y1 < 0 ? 0 : (y1 > H_ - 1 ? H_ - 1 : y1);

      // Coalesced 64B-per-half-wave gathers: lanes index consecutive channels.
      const float* p00 = vb + (size_t)(yc0 * W_ + xc0) * PIXSTRIDE;
      const float* p10 = vb + (size_t)(yc0 * W_ + xc1) * PIXSTRIDE;
      const float* p01 = vb + (size_t)(yc1 * W_ + xc0) * PIXSTRIDE;
      const float* p11 = vb + (size_t)(yc1 * W_ + xc1) * PIXSTRIDE;

      slo[j] = w00 * p00[n]      + w10 * p10[n]      + w01 * p01[n]      + w11 * p11[n];
      shi[j] = w00 * p00[n + 16] + w10 * p10[n + 16] + w01 * p01[n + 16] + w11 * p11[n + 16];
    }

    // A: only row t is nonzero this iteration (selector is branchless).
    const float sel = (n == t) ? 1.0f : 0.0f;
    v2f a;   a.x = sel * w2.x;  a.y = sel * w2.y;
    v2f blo; blo.x = slo[0];    blo.y = slo[1];
    v2f bhi; bhi.x = shi[0];    bhi.y = shi[1];

    dlo = __builtin_amdgcn_wmma_f32_16x16x4_f32(false, a, false, blo, (short)0, dlo, false, false);
    dhi = __builtin_amdgcn_wmma_f32_16x16x4_f32(false, a, false, bhi, (short)0, dhi, false, false);
  }

  // D layout: vgpr r -> (M = r + 8*hi, N = n).  out[b][q][h*HD + c]
  const int mb = hi * 8;
  #pragma unroll
  for (int r = 0; r < 8; ++r) {
    const int q = q0 + mb + r;
    float* o = out + ((size_t)b * NQ + q) * (NH * HD) + h * HD;
    o[n]      = dlo[r];
    o[n + 16] = dhi[r];
  }
}

extern "C" void kernel_launch(void* const* d_in, const int* in_sizes, int n_in,
                              void* d_out, int out_size, void* d_ws, size_t ws_size,
                              hipStream_t stream) {
  const float* value = (const float*)d_in[0];
  // d_in[1] = value_spatial_shapes (int64, fixed 100x100) — unused.
  const float* loc   = (const float*)d_in[2];
  const float* aw    = (const float*)d_in[3];
  float* out         = (float*)d_out;

  // 4*8*625 = 20000 wave-tiles, 8 waves (256 threads) per block -> 2500 blocks.
  deform_attn_wmma_kernel<<<dim3((BS * NH * NTQ) / 8), dim3(256), 0, stream>>>(
      value, loc, aw, out);
}